// MMCL_32289564131844
// MI455X (gfx1250) — compile-verified
//
#include <hip/hip_runtime.h>
#include <math.h>

// ---------------------------------------------------------------------------
// Hard-negative-mining contrastive loss, MI455X (gfx1250).
// One workgroup (1024 threads = 32 wave32) per row of 65536 fp32 logits.
//
// Pass A: stream row HBM -> LDS via CDNA5 async-to-LDS b128 loads, double-
//         buffered 16KB chunks with counted s_wait_asynccnt; build 2048-bin
//         radix histogram (digit = top 11 bits of order-preserving key) + max.
// Pass B: re-read row (L2-resident) and compact the boundary bucket into LDS;
//         refine remaining 21 bits in-LDS (11+10) -> exact Kth threshold.
// Pass C: re-read row (L2) accumulating exp(10x - m) over keys > threshold,
//         deterministic fixed-order reduction; + tie copies at threshold.
//
// LDS total ~62KB (safe under any CU-mode per-WG cap). HBM traffic ~1 pass:
// 256MB -> ~11us at 23.3 TB/s roofline; passes B/C hit the 192MB L2.
// ---------------------------------------------------------------------------

#define BLOCK   1024
#define BINS0   2048
#define CHUNKF  4096     // floats per chunk (16KB)
#define CAP_A   4096
#define CAP_B   1024

// scal[] slots (in LDS)
#define S_CNT   0   // compaction counter
#define S_KREM  1   // remaining K within current digit domain
#define S_THR   2   // threshold key prefix
#define S_VMAX  3   // row max key (target excluded)
#define S_NCAND 4   // candidate count for next round
#define S_POS   5   // positive logit bits
#define S_BSEL  6   // selected digit this round

__device__ __forceinline__ unsigned f2key(unsigned b) {
  // monotone map: larger float -> larger unsigned key
  return (b & 0x80000000u) ? ~b : (b | 0x80000000u);
}
__device__ __forceinline__ float key2f(unsigned k) {
  unsigned b = (k & 0x80000000u) ? (k ^ 0x80000000u) : ~k;
  return __uint_as_float(b);
}

// Find the digit bucket containing the krem-th largest element (descending).
__device__ __forceinline__ void find_bucket(const unsigned* hist, unsigned* sup,
                                            unsigned* scal, int bins, int tid) {
  const int ngroups = bins >> 5;
  if (tid < ngroups) {
    unsigned s = 0;
    for (int j = 0; j < 32; ++j) s += hist[(tid << 5) + j];
    sup[tid] = s;
  }
  __syncthreads();
  if (tid == 0) {
    unsigned krem = scal[S_KREM];
    unsigned cum = 0;
    int ssel = 0;
    for (int s = ngroups - 1; s >= 0; --s) {
      if (cum + sup[s] >= krem) { ssel = s; break; }
      cum += sup[s];
    }
    int bsel = ssel << 5;
    for (int b = (ssel << 5) + 31; b >= (ssel << 5); --b) {
      if (cum + hist[b] >= krem) { bsel = b; break; }
      cum += hist[b];
    }
    scal[S_KREM] = krem - cum;
    scal[S_BSEL] = (unsigned)bsel;
    scal[S_CNT]  = 0;
  }
  __syncthreads();
}

extern "C" __global__ void __launch_bounds__(BLOCK)
hnm_row_loss(const float* __restrict__ logits, const int* __restrict__ targets,
             float* __restrict__ row_loss, int N, int K) {
  extern __shared__ unsigned smem[];
  unsigned* buf0  = smem;               // CHUNKF words
  unsigned* buf1  = buf0 + CHUNKF;      // CHUNKF words
  unsigned* hist  = buf1 + CHUNKF;      // BINS0
  unsigned* candA = hist + BINS0;       // CAP_A (also fp reduce / max scratch)
  unsigned* candB = candA + CAP_A;      // CAP_B
  unsigned* sup   = candB + CAP_B;      // 64
  unsigned* scal  = sup + 64;           // 16

  const int tid = threadIdx.x;
  const float* row = logits + (size_t)blockIdx.x * (size_t)N;
  const int t = targets[blockIdx.x];    // uniform
  const int nchunks = N / CHUNKF;       // 16
  const int nvec = N >> 12;             // float4 iterations in passes B/C (16)
  const unsigned ldsbase = __builtin_amdgcn_groupstaticsize();

  // ---- init + prologue: issue chunk 0 async into buf0 ----
  {
    const int f = tid << 2;                               // chunk 0 floats
    unsigned lds_off = ldsbase + ((unsigned)(tid << 2) << 2);
    const float* gp = row + f;
    asm volatile("global_load_async_to_lds_b128 %0, %1, off"
                 :: "v"(lds_off), "v"(gp) : "memory");
  }
  for (int i = tid; i < BINS0; i += BLOCK) hist[i] = 0u;
  if (tid == 0) {
    scal[S_POS]  = __float_as_uint(row[t]);
    scal[S_KREM] = (unsigned)K;
    scal[S_THR]  = 0u;
    scal[S_CNT]  = 0u;
  }
  __syncthreads();

  // ---- Pass A: histogram digit0 + row max, async double-buffered ----
  unsigned lmax = 0u;
  for (int c = 0; c < nchunks; ++c) {
    if (c + 1 < nchunks) {
      const int f = ((c + 1) * CHUNKF) + (tid << 2);
      const unsigned bufw = ((c + 1) & 1) ? (unsigned)CHUNKF : 0u;
      unsigned lds_off = ldsbase + ((bufw + (unsigned)(tid << 2)) << 2);
      const float* gp = row + f;
      asm volatile("global_load_async_to_lds_b128 %0, %1, off"
                   :: "v"(lds_off), "v"(gp) : "memory");
#if __has_builtin(__builtin_amdgcn_s_wait_asynccnt)
      __builtin_amdgcn_s_wait_asynccnt(1);   // allow next chunk in flight
#else
      asm volatile("s_wait_asynccnt 0x1" ::: "memory");
#endif
    } else {
#if __has_builtin(__builtin_amdgcn_s_wait_asynccnt)
      __builtin_amdgcn_s_wait_asynccnt(0);
#else
      asm volatile("s_wait_asynccnt 0x0" ::: "memory");
#endif
    }
    __syncthreads();                         // chunk c visible to all waves
    const unsigned* cur = (c & 1) ? buf1 : buf0;
    #pragma unroll
    for (int j = 0; j < CHUNKF / BLOCK; ++j) {
      const int l = tid + (j << 10);
      const int g = c * CHUNKF + l;
      if (g == t) continue;                  // exclude positive column
      unsigned key = f2key(cur[l]);
      lmax = lmax > key ? lmax : key;
      atomicAdd(&hist[key >> 21], 1u);
    }
    __syncthreads();                         // buffer reusable for chunk c+2
  }
  candA[tid] = lmax;
  __syncthreads();
  for (int s = BLOCK / 2; s > 0; s >>= 1) {  // deterministic max-reduce
    if (tid < s) { unsigned a = candA[tid], b = candA[tid + s]; candA[tid] = a > b ? a : b; }
    __syncthreads();
  }
  if (tid == 0) scal[S_VMAX] = candA[0];
  __syncthreads();

  // ---- Round 0 bucket selection (top 11 bits) ----
  find_bucket(hist, sup, scal, BINS0, tid);
  if (tid == 0) scal[S_THR] |= scal[S_BSEL] << 21;
  __syncthreads();
  const unsigned b0 = scal[S_BSEL];

  // ---- Pass B: compact boundary bucket from global (L2-resident) ----
  for (int c = 0; c < nvec; ++c) {
    const int f = ((c << 10) + tid) << 2;
    const float4 v = *(const float4*)(row + f);
    const float e[4] = {v.x, v.y, v.z, v.w};
    #pragma unroll
    for (int k = 0; k < 4; ++k) {
      if (f + k == t) continue;
      unsigned key = f2key(__float_as_uint(e[k]));
      if ((key >> 21) == b0) {
        unsigned slot = atomicAdd(&scal[S_CNT], 1u);
        if (slot < CAP_A) candA[slot] = key;
      }
    }
  }
  __syncthreads();
  if (tid == 0) { unsigned c = scal[S_CNT]; scal[S_NCAND] = c < CAP_A ? c : CAP_A; scal[S_CNT] = 0; }
  __syncthreads();

  // ---- Round 1: next 11 bits over candidates (counts only) ----
  for (int i = tid; i < BINS0; i += BLOCK) hist[i] = 0u;
  __syncthreads();
  {
    const unsigned nA = scal[S_NCAND];
    for (unsigned i = tid; i < nA; i += BLOCK)
      atomicAdd(&hist[(candA[i] >> 10) & 2047u], 1u);
  }
  __syncthreads();
  find_bucket(hist, sup, scal, BINS0, tid);
  if (tid == 0) scal[S_THR] |= scal[S_BSEL] << 10;
  __syncthreads();
  {
    const unsigned nA = scal[S_NCAND];
    const unsigned bsel = scal[S_BSEL];
    for (unsigned i = tid; i < nA; i += BLOCK) {
      unsigned key = candA[i];
      if (((key >> 10) & 2047u) == bsel) {
        unsigned slot = atomicAdd(&scal[S_CNT], 1u);
        if (slot < CAP_B) candB[slot] = key;
      }
    }
  }
  __syncthreads();
  if (tid == 0) { unsigned c = scal[S_CNT]; scal[S_NCAND] = c < CAP_B ? c : CAP_B; scal[S_CNT] = 0; }
  __syncthreads();

  // ---- Round 2: last 10 bits ----
  for (int i = tid; i < 1024; i += BLOCK) hist[i] = 0u;
  __syncthreads();
  {
    const unsigned nB = scal[S_NCAND];
    for (unsigned i = tid; i < nB; i += BLOCK)
      atomicAdd(&hist[candB[i] & 1023u], 1u);
  }
  __syncthreads();
  find_bucket(hist, sup, scal, 1024, tid);
  if (tid == 0) scal[S_THR] |= scal[S_BSEL];
  __syncthreads();

  // ---- Pass C: exp-sum of exact top-K (fixed order, deterministic) ----
  {
    const unsigned thr  = scal[S_THR];
    const unsigned krem = scal[S_KREM];      // tie copies to include at thr
    const float pos  = __uint_as_float(scal[S_POS]);
    const float vmax = key2f(scal[S_VMAX]);
    const float m10  = 10.0f * fmaxf(pos, vmax);

    float s = 0.0f;
    for (int c = 0; c < nvec; ++c) {
      const int f = ((c << 10) + tid) << 2;
      const float4 v = *(const float4*)(row + f);
      const float e[4] = {v.x, v.y, v.z, v.w};
      #pragma unroll
      for (int k = 0; k < 4; ++k) {
        if (f + k == t) continue;
        if (f2key(__float_as_uint(e[k])) > thr)
          s += __expf(10.0f * e[k] - m10);
      }
    }
    float* fred = (float*)candA;             // candidates no longer needed
    fred[tid] = s;
    __syncthreads();
    for (int st = BLOCK / 2; st > 0; st >>= 1) {
      if (tid < st) fred[tid] += fred[tid + st];
      __syncthreads();
    }
    if (tid == 0) {
      float S   = fred[0] + (float)krem * __expf(10.0f * key2f(thr) - m10);
      float lse = m10 + __logf(2.0f * __expf(10.0f * pos - m10) + S);
      row_loss[blockIdx.x] = lse - 10.0f * pos;
    }
  }
}

extern "C" __global__ void __launch_bounds__(1024)
hnm_mean(const float* __restrict__ row_loss, float* __restrict__ out, int B) {
  __shared__ float red[1024];
  const int tid = threadIdx.x;
  float v = 0.0f;
  for (int i = tid; i < B; i += 1024) v += row_loss[i];
  red[tid] = v;
  __syncthreads();
  for (int s = 512; s > 0; s >>= 1) {
    if (tid < s) red[tid] += red[tid + s];
    __syncthreads();
  }
  if (tid == 0) out[0] = red[0] / (float)B;
}

extern "C" void kernel_launch(void* const* d_in, const int* in_sizes, int n_in,
                              void* d_out, int out_size, void* d_ws, size_t ws_size,
                              hipStream_t stream) {
  const float* logits  = (const float*)d_in[0];
  const int*   targets = (const int*)d_in[1];
  const int B = in_sizes[1];
  const int N = in_sizes[0] / B;
  const int K = (int)(0.01 * (double)(N - 1));   // 655 for N=65536

  float* row_loss = (float*)d_ws;                // B floats of scratch

  const size_t smem_bytes =
      (size_t)(2 * CHUNKF + BINS0 + CAP_A + CAP_B + 64 + 16) * sizeof(unsigned); // ~62KB

  (void)hipFuncSetAttribute((const void*)hnm_row_loss,
                            hipFuncAttributeMaxDynamicSharedMemorySize,
                            (int)smem_bytes);

  hipLaunchKernelGGL(hnm_row_loss, dim3(B), dim3(BLOCK), smem_bytes, stream,
                     logits, targets, row_loss, N, K);
  hipLaunchKernelGGL(hnm_mean, dim3(1), dim3(1024), 0, stream,
                     row_loss, (float*)d_out, B);
}